// NASNetController_22351009809191
// MI455X (gfx1250) — compile-verified
//
#include <hip/hip_runtime.h>
#include <hip/hip_bf16.h>

#define H       512
#define FOURH   2048
#define BATCH   4096
#define NOPS    8
#define TEMPR   5.0f
#define TANHC   2.5f

#define KPAD    516                 // padded K stride (516 % 64 == 4 -> conflict-free)
#define SX_OFF  0                   // x tile:  [16][KPAD]
#define SH_OFF  (16 * KPAD)         // h tile:  [16][KPAD]
#define SG_OFF  (32 * KPAD)         // gates:   [4][16][64]
#define SMEM_FLOATS (32 * KPAD + 4 * 16 * 64)

typedef __attribute__((ext_vector_type(2))) float v2f;
typedef __attribute__((ext_vector_type(8))) float v8f;

__device__ __forceinline__ float sigmoidf_(float x) { return 1.0f / (1.0f + expf(-x)); }

// ---------------------------------------------------------------------------
// Fused LSTM step.
// Block = 128 threads = 4 waves. Block tile: 16 batch rows x 64 hidden units,
// all 4 gates (wave w owns gate w, 4 N-subtiles of 16).
// Phase 1: stage x (embedding gather) and h tiles into LDS via CDNA5 async
//          copies (global_load_async_to_lds_b128, ASYNCcnt).
// Phase 2: K-loop: A-frags from LDS (ds_load_b64), B-frags (weights) from
//          global (L2-resident), 8x v_wmma_f32_16x16x4_f32 per k-step.
// Phase 3: gates through LDS, fused LSTM cell update.
// ---------------------------------------------------------------------------
__global__ __launch_bounds__(128)
void lstm_step_kernel(const float* __restrict__ emb,
                      const float* __restrict__ w_ih,
                      const float* __restrict__ w_hh,
                      const float* __restrict__ b_ih,
                      const float* __restrict__ b_hh,
                      const float* __restrict__ h_in,
                      float* __restrict__ h_out,
                      float* __restrict__ c,
                      const int* __restrict__ tokens)
{
    __shared__ float smem[SMEM_FLOATS];   // single shared array -> LDS offset 0

    const int wave = threadIdx.x >> 5;          // gate index: 0=i 1=f 2=g 3=o
    const int lane = threadIdx.x & 31;
    const int m0   = blockIdx.y << 4;           // batch-row tile (16 rows)
    const int jb   = blockIdx.x << 6;           // hidden-unit tile (64 units)
    const int l16  = lane & 15;

    // ---- Phase 1: async-stage A tiles (x, h) into LDS -----------------------
    // 16 rows x 512 floats each = 2048 b128 units per tile; 128 threads x 16.
    #pragma unroll 4
    for (int it = 0; it < 16; ++it) {
        const int u   = threadIdx.x + (it << 7);   // 0..2047
        const int row = u >> 7;                    // 0..15
        const int kc  = (u & 127) << 2;            // 0..508 step 4
        const int tok = tokens[m0 + row];
        const float* gx = emb  + (size_t)tok * H + kc;
        const float* gh = h_in + (size_t)(m0 + row) * H + kc;
        const unsigned lx = (unsigned)((SX_OFF + row * KPAD + kc) * 4);
        const unsigned lh = (unsigned)((SH_OFF + row * KPAD + kc) * 4);
        asm volatile("global_load_async_to_lds_b128 %0, %1, off"
                     :: "v"(lx), "v"(gx) : "memory");
        asm volatile("global_load_async_to_lds_b128 %0, %1, off"
                     :: "v"(lh), "v"(gh) : "memory");
    }
    asm volatile("s_wait_asynccnt 0x0" ::: "memory");
    __syncthreads();

    // ---- Phase 2: K-loop with WMMA -----------------------------------------
    // A-frag (16x4 f32): lanes 0-15 row M=l16 hold K=0(V0),1(V1); lanes 16-31
    // hold K=2,3.  B-frag (4x16): lane gives N=l16, same K split.
    const int koff = (lane < 16) ? 0 : 2;

    const float* __restrict__ wih0 = w_ih + (size_t)(wave * H + jb +  0 + l16) * H;
    const float* __restrict__ wih1 = w_ih + (size_t)(wave * H + jb + 16 + l16) * H;
    const float* __restrict__ wih2 = w_ih + (size_t)(wave * H + jb + 32 + l16) * H;
    const float* __restrict__ wih3 = w_ih + (size_t)(wave * H + jb + 48 + l16) * H;
    const float* __restrict__ whh0 = w_hh + (size_t)(wave * H + jb +  0 + l16) * H;
    const float* __restrict__ whh1 = w_hh + (size_t)(wave * H + jb + 16 + l16) * H;
    const float* __restrict__ whh2 = w_hh + (size_t)(wave * H + jb + 32 + l16) * H;
    const float* __restrict__ whh3 = w_hh + (size_t)(wave * H + jb + 48 + l16) * H;

    const float* ax_base = smem + SX_OFF + l16 * KPAD + koff;
    const float* ah_base = smem + SH_OFF + l16 * KPAD + koff;

    v8f a0 = {}, a1 = {}, a2 = {}, a3 = {};
    #pragma unroll 2
    for (int k = 0; k < H; k += 4) {
        const v2f ax = *(const v2f*)(ax_base + k);
        const v2f ah = *(const v2f*)(ah_base + k);
        const v2f bi0 = *(const v2f*)(wih0 + k + koff);
        const v2f bi1 = *(const v2f*)(wih1 + k + koff);
        const v2f bi2 = *(const v2f*)(wih2 + k + koff);
        const v2f bi3 = *(const v2f*)(wih3 + k + koff);
        const v2f bh0 = *(const v2f*)(whh0 + k + koff);
        const v2f bh1 = *(const v2f*)(whh1 + k + koff);
        const v2f bh2 = *(const v2f*)(whh2 + k + koff);
        const v2f bh3 = *(const v2f*)(whh3 + k + koff);
        a0 = __builtin_amdgcn_wmma_f32_16x16x4_f32(false, ax, false, bi0, (short)0, a0, false, false);
        a1 = __builtin_amdgcn_wmma_f32_16x16x4_f32(false, ax, false, bi1, (short)0, a1, false, false);
        a2 = __builtin_amdgcn_wmma_f32_16x16x4_f32(false, ax, false, bi2, (short)0, a2, false, false);
        a3 = __builtin_amdgcn_wmma_f32_16x16x4_f32(false, ax, false, bi3, (short)0, a3, false, false);
        a0 = __builtin_amdgcn_wmma_f32_16x16x4_f32(false, ah, false, bh0, (short)0, a0, false, false);
        a1 = __builtin_amdgcn_wmma_f32_16x16x4_f32(false, ah, false, bh1, (short)0, a1, false, false);
        a2 = __builtin_amdgcn_wmma_f32_16x16x4_f32(false, ah, false, bh2, (short)0, a2, false, false);
        a3 = __builtin_amdgcn_wmma_f32_16x16x4_f32(false, ah, false, bh3, (short)0, a3, false, false);
    }

    // ---- Phase 3: bias + gate staging + LSTM cell update --------------------
    const int mbase = (lane < 16) ? 0 : 8;      // C/D layout: VGPR v -> M=v(+8)
    {
        v8f accs[4] = { a0, a1, a2, a3 };
        #pragma unroll
        for (int n = 0; n < 4; ++n) {
            const int ncol = wave * H + jb + n * 16 + l16;
            const float bias = b_ih[ncol] + b_hh[ncol];
            float* sg = smem + SG_OFF + (wave * 16) * 64 + n * 16 + l16;
            #pragma unroll
            for (int v = 0; v < 8; ++v)
                sg[(mbase + v) * 64] = accs[n][v] + bias;
        }
    }
    __syncthreads();

    #pragma unroll
    for (int e = threadIdx.x; e < 1024; e += 128) {
        const int m = e >> 6, j = e & 63;
        const float ig = smem[SG_OFF + (0 * 16 + m) * 64 + j];
        const float fg = smem[SG_OFF + (1 * 16 + m) * 64 + j];
        const float gg = smem[SG_OFF + (2 * 16 + m) * 64 + j];
        const float og = smem[SG_OFF + (3 * 16 + m) * 64 + j];
        const size_t idx = (size_t)(m0 + m) * H + (jb + j);
        const float cn = sigmoidf_(fg) * c[idx] + sigmoidf_(ig) * tanhf(gg);
        c[idx]     = cn;
        h_out[idx] = sigmoidf_(og) * tanhf(cn);
    }
}

// ---------------------------------------------------------------------------
// Decision kernel: one wave per batch row (unchanged from round 1).
// ---------------------------------------------------------------------------
__device__ __forceinline__ void decide_(const float* lg, int k, const float* g,
                                        int& action, float& sel, float& ent)
{
    float mx = lg[0];
    for (int j = 1; j < k; ++j) mx = fmaxf(mx, lg[j]);
    float se = 0.0f;
    for (int j = 0; j < k; ++j) se += expf(lg[j] - mx);
    const float logZ = mx + logf(se);
    float best = -1e30f; int a = 0; float e = 0.0f;
    for (int j = 0; j < k; ++j) {
        const float lp = lg[j] - logZ;
        e -= lp * expf(lp);
        const float sc = lg[j] + g[j];
        if (sc > best) { best = sc; a = j; }   // strict '>' == first-max argmax
    }
    action = a; sel = lg[a] - logZ; ent = e;
}

__global__ __launch_bounds__(256)
void decide_kernel(const float* __restrict__ h,
                   const float* __restrict__ wn, const float* __restrict__ bn, int knode,
                   const float* __restrict__ wop, const float* __restrict__ bop,
                   const float* __restrict__ gum_node, const float* __restrict__ gum_op,
                   int* __restrict__ act_node, float* __restrict__ lp_node, float* __restrict__ ent_node,
                   int* __restrict__ act_op,   float* __restrict__ lp_op,   float* __restrict__ ent_op,
                   int* __restrict__ tokens)
{
    const int wave = threadIdx.x >> 5;
    const int lane = threadIdx.x & 31;
    const int b    = blockIdx.x * 8 + wave;

    const float* __restrict__ hrow = h + (size_t)b * H;
    float hv[16];
    #pragma unroll
    for (int t = 0; t < 16; ++t) hv[t] = hrow[lane + 32 * t];

    float lg[8];

    // node-predecessor logits: TANH_C * tanh((h @ wn^T + bn) / TEMP)
    for (int j = 0; j < knode; ++j) {
        const float* __restrict__ w = wn + (size_t)j * H;
        float s = 0.0f;
        #pragma unroll
        for (int t = 0; t < 16; ++t) s += hv[t] * w[lane + 32 * t];
        #pragma unroll
        for (int off = 16; off; off >>= 1) s += __shfl_xor(s, off, 32);
        lg[j] = TANHC * tanhf((s + bn[j]) / TEMPR);
    }
    {
        int a; float sel, ent;
        const float* g = gum_node + (size_t)b * NOPS;
        float gl[8];
        for (int j = 0; j < knode; ++j) gl[j] = g[j];
        decide_(lg, knode, gl, a, sel, ent);
        if (lane == 0) { act_node[b] = a; lp_node[b] = sel; ent_node[b] = ent; }
    }

    // op logits: (h @ wop^T + bop) / TEMP
    #pragma unroll
    for (int j = 0; j < NOPS; ++j) {
        const float* __restrict__ w = wop + (size_t)j * H;
        float s = 0.0f;
        #pragma unroll
        for (int t = 0; t < 16; ++t) s += hv[t] * w[lane + 32 * t];
        #pragma unroll
        for (int off = 16; off; off >>= 1) s += __shfl_xor(s, off, 32);
        lg[j] = (s + bop[j]) / TEMPR;
    }
    {
        int a; float sel, ent;
        const float* g = gum_op + (size_t)b * NOPS;
        float gl[8];
        #pragma unroll
        for (int j = 0; j < NOPS; ++j) gl[j] = g[j];
        decide_(lg, NOPS, gl, a, sel, ent);
        if (lane == 0) {
            act_op[b] = a; lp_op[b] = sel; ent_op[b] = ent;
            tokens[b] = a;      // op action feeds next embedding lookup
        }
    }
}

__global__ void init_kernel(float* __restrict__ h0, float* __restrict__ cc,
                            int* __restrict__ tokens)
{
    const int i = blockIdx.x * blockDim.x + threadIdx.x;
    if (i < BATCH * H) { h0[i] = 0.0f; cc[i] = 0.0f; }
    if (i < BATCH) tokens[i] = 0;
}

extern "C" void kernel_launch(void* const* d_in, const int* in_sizes, int n_in,
                              void* d_out, int out_size, void* d_ws, size_t ws_size,
                              hipStream_t stream)
{
    (void)in_sizes; (void)n_in; (void)out_size; (void)ws_size;

    const float* emb   = (const float*)d_in[0];
    const float* w_ih  = (const float*)d_in[1];
    const float* w_hh  = (const float*)d_in[2];
    const float* b_ih  = (const float*)d_in[3];
    const float* b_hh  = (const float*)d_in[4];
    const float* wn[4] = { (const float*)d_in[5], (const float*)d_in[7],
                           (const float*)d_in[9], (const float*)d_in[11] };
    const float* bn[4] = { (const float*)d_in[6], (const float*)d_in[8],
                           (const float*)d_in[10], (const float*)d_in[12] };
    const float* wop    = (const float*)d_in[13];
    const float* bop    = (const float*)d_in[14];
    const float* gumbel = (const float*)d_in[15];

    // workspace: h double-buffer + c + tokens (~25 MB)
    float* ws  = (float*)d_ws;
    float* h0  = ws;
    float* h1  = ws + (size_t)BATCH * H;
    float* cc  = ws + 2 * (size_t)BATCH * H;
    int* tokens = (int*)(ws + 3 * (size_t)BATCH * H);

    // outputs: [16,B] int32 actions, [16,B] f32 log_probs, [16,B] f32 entropies
    int*   acts = (int*)d_out;
    float* lps  = (float*)d_out + 16 * (size_t)BATCH;
    float* ents = (float*)d_out + 32 * (size_t)BATCH;

    init_kernel<<<(BATCH * H + 255) / 256, 256, 0, stream>>>(h0, cc, tokens);

    for (int step = 0; step < 8; ++step) {
        const int node = step >> 1;
        float* hin  = (step & 1) ? h1 : h0;
        float* hout = (step & 1) ? h0 : h1;

        lstm_step_kernel<<<dim3(H / 64, BATCH / 16), 128, 0, stream>>>(
            emb, w_ih, w_hh, b_ih, b_hh, hin, hout, cc, tokens);

        const int sN = 2 * step, sO = 2 * step + 1;
        decide_kernel<<<BATCH / 8, 256, 0, stream>>>(
            hout,
            wn[node], bn[node], node + 2,
            wop + (size_t)node * NOPS * H, bop + (size_t)node * NOPS,
            gumbel + (size_t)sN * BATCH * NOPS, gumbel + (size_t)sO * BATCH * NOPS,
            acts + (size_t)sN * BATCH, lps + (size_t)sN * BATCH, ents + (size_t)sN * BATCH,
            acts + (size_t)sO * BATCH, lps + (size_t)sO * BATCH, ents + (size_t)sO * BATCH,
            tokens);
    }
}